// ChamferDistLoss_80676665688371
// MI455X (gfx1250) — compile-verified
//
#include <hip/hip_runtime.h>

typedef __attribute__((ext_vector_type(2))) float v2f;
typedef __attribute__((ext_vector_type(4))) float v4f;
typedef __attribute__((ext_vector_type(8))) float v8f;

#define WAVES_PER_BLOCK 8
#define BLOCK_THREADS   (WAVES_PER_BLOCK * 32)
#define CHUNKS          8      // column chunks per side (occupancy: ~8 waves/SIMD)
#define GROUP           4      // tiles per software-pipeline stage

static __device__ __forceinline__ v8f wmma_tile(v2f a, v2f b, v8f c) {
    return __builtin_amdgcn_wmma_f32_16x16x4_f32(
        /*neg_a=*/false, a, /*neg_b=*/false, b,
        /*c_mod=*/(short)0, c, /*reuse_a=*/false, /*reuse_b=*/false);
}

// Clear accumulator and set all per-row min slots to +inf (ws is poisoned with
// 0xAA once by the harness and never re-poisoned, so do this every call).
__global__ __launch_bounds__(256)
void chamfer_init_kernel(float* accum, unsigned int* rowBits, int total) {
    const int i = blockIdx.x * blockDim.x + threadIdx.x;
    if (i == 0) accum[0] = 0.0f;
    for (int k = i; k < total; k += gridDim.x * blockDim.x)
        rowBits[k] = 0x7F800000u;                 // +inf
}

// Precompute augmented operand: dst[i] = [x, y, z, |pt|^2] (float4).
// A lane's WMMA B-operand is then just 8 contiguous bytes of this record:
//   lanes 0-15 (K0,K1) -> bytes [0,8) = (x, y)
//   lanes 16-31 (K2,K3) -> bytes [8,16) = (z, |pt|^2)
__global__ __launch_bounds__(256)
void chamfer_augment_kernel(const float* __restrict__ src,
                            v4f* __restrict__ dst, int n) {
    const int i = blockIdx.x * blockDim.x + threadIdx.x;
    if (i < n) {
        const float x = src[3 * i + 0];
        const float y = src[3 * i + 1];
        const float z = src[3 * i + 2];
        v4f o;
        o.x = x; o.y = y; o.z = z;
        o.w = fmaf(x, x, fmaf(y, y, z * z));
        dst[i] = o;                               // global_store_b128
    }
}

// One (row-tile, column-chunk) pair per wave.  D = A*B + C gives the full
// 16x16 squared-distance tile in ONE v_wmma_f32_16x16x4_f32:
//   A row = [-2px, -2py, -2pz, 1],  B col = [tx, ty, tz, |t|^2],  C = |p|^2.
// Ping-pong software pipeline (no register rotation), scalar loop control via
// readfirstlane (EXEC stays all-1s at every WMMA, as the ISA requires).
// Per-chunk row mins are clamped then published with global_atomic_min_u32
// (bit order == numeric order for floats >= 0).
__global__ __launch_bounds__(BLOCK_THREADS)
void chamfer_oneside_kernel(const v4f* __restrict__ Aaug,
                            const v4f* __restrict__ Baug,
                            unsigned int* __restrict__ rowBits,
                            int n, int m) {
    const int lane = threadIdx.x & 31;
    const int waveV = blockIdx.x * WAVES_PER_BLOCK + (threadIdx.x >> 5);
    const int wave  = __builtin_amdgcn_readfirstlane(waveV);   // wave-uniform SGPR
    const int ntiles = n >> 4;
    const int mtiles = m >> 4;
    if (wave >= ntiles * CHUNKS) return;

    const int chunk = wave / ntiles;      // 8 waves of a block share one chunk
    const int rtile = wave % ntiles;      // ...and take consecutive row tiles
    const int half  = lane >> 4;          // 0: K=0,1 ; 1: K=2,3
    const int sub   = lane & 15;

    // ---- A tile: 16 predict points (one augmented b128 per lane).
    const v4f p = Aaug[rtile * 16 + sub];

    v2f a;
    a.x = half ? (-2.0f * p.z) : (-2.0f * p.x);   // K=2 : K=0
    a.y = half ? 1.0f          : (-2.0f * p.y);   // K=3 : K=1

    // ---- C tile: |p|^2 per row in the C/D layout (VGPR v -> rows v, v+8).
    v8f cmat;
    #pragma unroll
    for (int v = 0; v < 8; ++v)
        cmat[v] = __shfl(p.w, v + 8 * half, 32);

    v8f rowmin;
    #pragma unroll
    for (int v = 0; v < 8; ++v) rowmin[v] = 3.0e38f;

    // ---- This wave's column range (all scalar).
    const int cpc   = (mtiles + CHUNKS - 1) / CHUNKS;
    const int ct0   = chunk * cpc;
    const int ctEnd = (ct0 + cpc < mtiles) ? (ct0 + cpc) : mtiles;
    const int count = ctEnd - ct0;
    if (count <= 0) return;

    const char* bp = (const char*)Baug + 256u * (unsigned)ct0
                   + (unsigned)(16 * sub + 8 * half);

    const int ngroups = count / GROUP;
    const int npairs  = ngroups >> 1;             // pipeline advances 2 groups/iter
    const int xgroup  = ngroups & 1;              // leftover whole group
    const int rem     = count - ngroups * GROUP;  // leftover tiles (< GROUP)

    v2f b0[GROUP], b1[GROUP];
    long off = 0;
    const long GB = 256 * GROUP;

    #define LOADG(buf, o)                                                     \
        _Pragma("unroll")                                                     \
        for (int u = 0; u < GROUP; ++u)                                       \
            (buf)[u] = *(const v2f*)(bp + (o) + 256 * u);
    #define COMPG(buf)                                                        \
        _Pragma("unroll")                                                     \
        for (int u = 0; u < GROUP; ++u) {                                     \
            v8f d = wmma_tile(a, (buf)[u], cmat);                             \
            _Pragma("unroll")                                                 \
            for (int v = 0; v < 8; ++v)                                       \
                rowmin[v] = fminf(rowmin[v], d[v]); /* fuses into v_min3 */   \
        }

    if (npairs > 0) {
        LOADG(b0, off);
        for (int pr = 0; pr < npairs - 1; ++pr) {   // scalar-controlled loop
            LOADG(b1, off + GB);                    // prefetch group g+1
            COMPG(b0);                              // compute group g
            LOADG(b0, off + 2 * GB);                // prefetch group g+2
            COMPG(b1);                              // compute group g+1
            off += 2 * GB;
        }
        LOADG(b1, off + GB);                        // peeled last pair: no
        COMPG(b0);                                  // out-of-range prefetch
        COMPG(b1);
        off += 2 * GB;
    }
    if (xgroup) {
        LOADG(b0, off);
        COMPG(b0);
        off += GB;
    }
    for (int t = 0; t < rem; ++t) {                 // generic remainder
        const v2f b = *(const v2f*)(bp + off + 256 * t);
        v8f d = wmma_tile(a, b, cmat);
        #pragma unroll
        for (int v = 0; v < 8; ++v)
            rowmin[v] = fminf(rowmin[v], d[v]);
    }
    #undef LOADG
    #undef COMPG

    // ---- Row-min across the 16 lanes of each half, clamp (commutes with
    //      min), then publish per-row partial mins as uint atomic-min.
    #pragma unroll
    for (int v = 0; v < 8; ++v) {
        float r = rowmin[v];
        r = fminf(r, __shfl_xor(r, 1, 32));
        r = fminf(r, __shfl_xor(r, 2, 32));
        r = fminf(r, __shfl_xor(r, 4, 32));
        r = fminf(r, __shfl_xor(r, 8, 32));
        r = fmaxf(r, 0.0f);
        if (sub == v)                               // one lane per half
            atomicMin(&rowBits[rtile * 16 + v + 8 * half], __float_as_uint(r));
    }
}

// Sum clamped per-row minima of both directions: accum = sum(d1)/N + sum(d2)/M.
__global__ __launch_bounds__(256)
void chamfer_reduce_kernel(const unsigned int* __restrict__ bits,
                           int n, int m, float invN, float invM,
                           float* __restrict__ accum) {
    __shared__ float red[256];
    const int total = n + m;
    float s = 0.0f;
    for (int i = blockIdx.x * blockDim.x + threadIdx.x; i < total;
         i += gridDim.x * blockDim.x)
        s += __uint_as_float(bits[i]) * (i < n ? invN : invM);
    red[threadIdx.x] = s;
    __syncthreads();
    for (int off = 128; off > 0; off >>= 1) {
        if (threadIdx.x < off) red[threadIdx.x] += red[threadIdx.x + off];
        __syncthreads();
    }
    if (threadIdx.x == 0) atomicAdd(accum, red[0]);
}

__global__ void chamfer_finalize_kernel(const float* __restrict__ accum,
                                        float* __restrict__ out) {
    if (threadIdx.x == 0 && blockIdx.x == 0) out[0] = accum[0];
}

extern "C" void kernel_launch(void* const* d_in, const int* in_sizes, int n_in,
                              void* d_out, int out_size, void* d_ws, size_t ws_size,
                              hipStream_t stream) {
    const float* predict = (const float*)d_in[0];   // [1, N, 3] f32
    const float* target  = (const float*)d_in[1];   // [1, M, 3] f32
    const int N = in_sizes[0] / 3;
    const int M = in_sizes[1] / 3;

    // ws layout: [0,16)                       accumulator (float + pad)
    //            [16, 16+4(N+M))              per-row min bits (dir1 | dir2)
    //            [16+4(N+M), ... +16(N+M))    augmented P then augmented T
    float*        accum   = (float*)d_ws;
    unsigned int* rowBits = (unsigned int*)((char*)d_ws + 16);
    v4f*          augP    = (v4f*)((char*)d_ws + 16 + (size_t)4 * (N + M));
    v4f*          augT    = augP + N;
    float*        out     = (float*)d_out;

    chamfer_init_kernel<<<64, 256, 0, stream>>>(accum, rowBits, N + M);
    chamfer_augment_kernel<<<(N + 255) / 256, 256, 0, stream>>>(predict, augP, N);
    chamfer_augment_kernel<<<(M + 255) / 256, 256, 0, stream>>>(target,  augT, M);

    const int wavesP = (N >> 4) * CHUNKS;
    const int wavesT = (M >> 4) * CHUNKS;
    const int gridP = (wavesP + WAVES_PER_BLOCK - 1) / WAVES_PER_BLOCK;
    const int gridT = (wavesT + WAVES_PER_BLOCK - 1) / WAVES_PER_BLOCK;

    // dist1: nearest target for every predict point
    chamfer_oneside_kernel<<<gridP, BLOCK_THREADS, 0, stream>>>(
        augP, augT, rowBits, N, M);
    // dist2: nearest predict for every target point (symmetric, args swapped)
    chamfer_oneside_kernel<<<gridT, BLOCK_THREADS, 0, stream>>>(
        augT, augP, rowBits + N, M, N);

    chamfer_reduce_kernel<<<64, 256, 0, stream>>>(
        rowBits, N, M, 1.0f / (float)N, 1.0f / (float)M, accum);
    chamfer_finalize_kernel<<<1, 32, 0, stream>>>(accum, out);
}